// ListMLELoss_84112639525733
// MI455X (gfx1250) — compile-verified
//
#include <hip/hip_runtime.h>
#include <math.h>

#define NROW      8192
#define BATCH     2048
#define NTHREADS  1024
#define NWAVES    (NTHREADS / 32)
#define TILE      256
// LDS: 8192 u64 keys (64KB, overlaid by 8192-float t-array) + 8192 f32 scores (32KB) + 64 f32 scratch
#define SHMEM_BYTES (NROW * 8 + NROW * 4 + 64 * 4)

typedef float        v2f __attribute__((ext_vector_type(2)));
typedef float        v8f __attribute__((ext_vector_type(8)));
typedef unsigned int v4u __attribute__((ext_vector_type(4)));
typedef int          v4i __attribute__((ext_vector_type(4)));
typedef int          v8i __attribute__((ext_vector_type(8)));

#if defined(__gfx1250__) && __has_builtin(__builtin_amdgcn_wmma_f32_16x16x4_f32)
#define USE_WMMA 1
#else
#define USE_WMMA 0
#endif

#if defined(__gfx1250__) && __has_builtin(__builtin_amdgcn_tensor_load_to_lds) && \
    __has_builtin(__builtin_amdgcn_s_wait_tensorcnt)
#define USE_TDM 1
#else
#define USE_TDM 0
#endif

// Compile-time path confirmation (device pass only; silence == CDNA5 path active).
#if defined(__gfx1250__) && !USE_TDM
#warning "CDNA5: tensor_load_to_lds builtin unavailable -> global-load staging fallback"
#endif
#if defined(__gfx1250__) && !USE_WMMA
#warning "CDNA5: wmma_f32_16x16x4_f32 builtin unavailable -> VALU scan fallback"
#endif

#if USE_TDM
// Build a minimal 2D D# (1 row of `nelem` 4-byte elements) and issue TENSOR_LOAD_TO_LDS.
__device__ __forceinline__ void tdm_load_row(const void* gsrc, unsigned lds_byte_off,
                                             unsigned nelem) {
  unsigned long long ga = (unsigned long long)(uintptr_t)gsrc;
  v4u g0;
  g0.x = 1u;                                                  // count=1 (valid), user mode
  g0.y = lds_byte_off;                                        // lds_addr
  g0.z = (unsigned)(ga & 0xFFFFFFFFu);                        // global_addr[31:0]
  g0.w = (unsigned)((ga >> 32) & 0x01FFFFFFu) | (2u << 30);   // global_addr[56:32], type=2
  v8i g1;
  g1[0] = (int)(2u << 16);                          // wg_mask=0, data_size=2 (4 bytes)
  g1[1] = (int)((nelem & 0xFFFFu) << 16);           // tensor_dim0[15:0] at bits[63:48]
  g1[2] = (int)((nelem >> 16) & 0xFFFFu) | (int)(1u << 16);  // dim0[31:16], tensor_dim1=1
  g1[3] = (int)((nelem & 0xFFFFu) << 16);           // tile_dim0 = nelem
  g1[4] = 1;                                        // tile_dim1=1, tile_dim2=0
  g1[5] = (int)nelem;                               // tensor_dim0_stride[31:0]
  g1[6] = (int)((nelem & 0xFFFFu) << 16);           // stride0[47:32]=0, stride1[15:0]
  g1[7] = 0;                                        // stride1[47:16]=0
  v4i z4 = {0, 0, 0, 0};
#if __clang_major__ >= 23
  v8i z8 = {0, 0, 0, 0, 0, 0, 0, 0};
  __builtin_amdgcn_tensor_load_to_lds(g0, g1, z4, z4, z8, 0);
#else
  __builtin_amdgcn_tensor_load_to_lds(g0, g1, z4, z4, 0);
#endif
}
#endif

// In-register bitonic compare-exchange (equal keys: order irrelevant, key is unique anyway).
__device__ __forceinline__ void ce64(unsigned long long& a, unsigned long long& b,
                                     bool asc) {
  unsigned long long lo = (a < b) ? a : b;
  unsigned long long hi = (a < b) ? b : a;
  a = asc ? lo : hi;
  b = asc ? hi : lo;
}

__global__ __launch_bounds__(NTHREADS) void listmle_row_kernel(
    const float* __restrict__ scores, const float* __restrict__ labels,
    float* __restrict__ rowsums) {
  extern __shared__ __align__(16) unsigned char smem[];
  unsigned long long* lds_keys = (unsigned long long*)smem;          // 64KB
  float* lds_t    = (float*)smem;                                    // overlays keys
  float* lds_scr  = (float*)(smem + NROW * 8);                       // 32KB
  float* lds_tot  = (float*)(smem + NROW * 8 + NROW * 4);            // 32 floats
  float* lds_wsum = lds_tot + NWAVES;                                // 32 floats

  const int tid  = threadIdx.x;
  const int row  = blockIdx.x;
  const int base = tid * 8;  // each thread owns 8 contiguous elements (64B aligned in keys)
  const float* slab = labels + (size_t)row * NROW;
  const float* ssc  = scores + (size_t)row * NROW;

  // ---- Stage labels (into t-array slots as floats) and scores into LDS ----
#if USE_TDM
  if (tid == 0) {
    unsigned lb = (unsigned)(unsigned long long)(uintptr_t)smem;  // LDS byte offset
    tdm_load_row((const void*)slab, lb, NROW);
    tdm_load_row((const void*)ssc, lb + NROW * 8, NROW);
    __builtin_amdgcn_s_wait_tensorcnt(0);
  }
  __syncthreads();
#else
  for (int i = tid; i < NROW; i += NTHREADS) {
    lds_t[i]   = slab[i];
    lds_scr[i] = ssc[i];
  }
  __syncthreads();
#endif

  // ---- Phase 1: build keys in registers + in-register bitonic stages k=2,4,8 ----
  // key = (~monotonic(label)) << 32 | index : ascending u64 == stable descending label.
  {
    float lv[8];
#pragma unroll
    for (int q = 0; q < 8; ++q) lv[q] = lds_t[base + q];
    __syncthreads();  // all label reads done before keys overwrite the region

    unsigned long long kr[8];
#pragma unroll
    for (int q = 0; q < 8; ++q) {
      unsigned u = __float_as_uint(lv[q]);
      unsigned m = (u & 0x80000000u) ? ~u : (u | 0x80000000u);  // monotonic ascending map
      kr[q] = ((unsigned long long)(~m) << 32) | (unsigned)(base + q);
    }
    // stage k=2 (j=1): dir = ((q&2)==0)
    ce64(kr[0], kr[1], true);  ce64(kr[2], kr[3], false);
    ce64(kr[4], kr[5], true);  ce64(kr[6], kr[7], false);
    // stage k=4 (j=2,1): dir = ((q&4)==0)
    ce64(kr[0], kr[2], true);  ce64(kr[1], kr[3], true);
    ce64(kr[4], kr[6], false); ce64(kr[5], kr[7], false);
    ce64(kr[0], kr[1], true);  ce64(kr[2], kr[3], true);
    ce64(kr[4], kr[5], false); ce64(kr[6], kr[7], false);
    // stage k=8 (j=4,2,1): dir uniform per thread = ((tid&1)==0)
    {
      const bool a8 = ((tid & 1) == 0);
      ce64(kr[0], kr[4], a8); ce64(kr[1], kr[5], a8);
      ce64(kr[2], kr[6], a8); ce64(kr[3], kr[7], a8);
      ce64(kr[0], kr[2], a8); ce64(kr[1], kr[3], a8);
      ce64(kr[4], kr[6], a8); ce64(kr[5], kr[7], a8);
      ce64(kr[0], kr[1], a8); ce64(kr[2], kr[3], a8);
      ce64(kr[4], kr[5], a8); ce64(kr[6], kr[7], a8);
    }
#pragma unroll
    for (int q = 0; q < 8; ++q) lds_keys[base + q] = kr[q];
  }
  __syncthreads();

  // ---- Stages k=16..8192: LDS passes for j>=8, register tail for j=4,2,1 ----
  // Final stage fuses the gather: sorted keys stay in VGPRs, t=exp(score[idx]) directly.
  for (unsigned k = 16; k <= NROW; k <<= 1) {
    for (unsigned j = k >> 1; j >= 8; j >>= 1) {
      for (unsigned p0 = (unsigned)tid; p0 < NROW / 2; p0 += NTHREADS) {
        unsigned i = ((p0 & ~(j - 1u)) << 1) | (p0 & (j - 1u));
        unsigned l = i | j;
        unsigned long long a = lds_keys[i];
        unsigned long long b = lds_keys[l];
        bool asc = ((i & k) == 0u);
        if ((a > b) == asc) {
          lds_keys[i] = b;
          lds_keys[l] = a;
        }
      }
      __syncthreads();
    }
    // register tail: j=4,2,1 stay inside this thread's aligned 8-block; dir uniform
    unsigned long long kr[8];
#pragma unroll
    for (int q = 0; q < 8; ++q) kr[q] = lds_keys[base + q];
    {
      const bool asc = (((unsigned)base & k) == 0u);
      ce64(kr[0], kr[4], asc); ce64(kr[1], kr[5], asc);
      ce64(kr[2], kr[6], asc); ce64(kr[3], kr[7], asc);
      ce64(kr[0], kr[2], asc); ce64(kr[1], kr[3], asc);
      ce64(kr[4], kr[6], asc); ce64(kr[5], kr[7], asc);
      ce64(kr[0], kr[1], asc); ce64(kr[2], kr[3], asc);
      ce64(kr[4], kr[5], asc); ce64(kr[6], kr[7], asc);
    }
    if (k < NROW) {
      #pragma unroll
      for (int q = 0; q < 8; ++q) lds_keys[base + q] = kr[q];
      __syncthreads();
    } else {
      // final stage: gather scores straight from register keys, t = exp(s)
      float tvv[8];
#pragma unroll
      for (int q = 0; q < 8; ++q) {
        unsigned idx = (unsigned)(kr[q] & 0xFFFFFFFFu);
        tvv[q] = expf(lds_scr[idx]);
      }
      __syncthreads();  // all tail key-loads complete before t overwrites key region
#pragma unroll
      for (int q = 0; q < 8; ++q) lds_t[base + q] = tvv[q];
    }
  }
  __syncthreads();

  // ---- Per-wave 256-element inclusive scan of one tile, viewed as 16x16 matrix T ----
  // P = T*U_incl + rowsum(L_strict*T); U/L generated per-lane in native B/A layouts.
  const int wave = tid >> 5;
  const int lane = tid & 31;
  const int half = lane >> 4;
  const int n    = lane & 15;
  const int tb   = wave * TILE;

  float p[8];
#if USE_WMMA
  {
    v8f x = {0.f, 0.f, 0.f, 0.f, 0.f, 0.f, 0.f, 0.f};
    v8f z = {0.f, 0.f, 0.f, 0.f, 0.f, 0.f, 0.f, 0.f};
#pragma unroll
    for (int kk = 0; kk < 4; ++kk) {
      const int k0 = 4 * kk + 2 * half;
      // X += T(:, k0..k0+3) * U(k0..k0+3, :)
      v2f a, bu;
      a.x  = lds_t[tb + 16 * n + k0];
      a.y  = lds_t[tb + 16 * n + k0 + 1];
      bu.x = (k0     <= n) ? 1.0f : 0.0f;
      bu.y = (k0 + 1 <= n) ? 1.0f : 0.0f;
      x = __builtin_amdgcn_wmma_f32_16x16x4_f32(false, a, false, bu, (short)0, x,
                                                false, false);
      // Z += L(:, k0..k0+3) * T(k0..k0+3, :)
      v2f al, bt;
      al.x = (k0     < n) ? 1.0f : 0.0f;
      al.y = (k0 + 1 < n) ? 1.0f : 0.0f;
      bt.x = lds_t[tb + 16 * k0 + n];
      bt.y = lds_t[tb + 16 * (k0 + 1) + n];
      z = __builtin_amdgcn_wmma_f32_16x16x4_f32(false, al, false, bt, (short)0, z,
                                                false, false);
    }
#pragma unroll
    for (int j = 0; j < 8; ++j) {
      float rs = z[j];  // row (j + 8*half) of Z across 16 lanes of this half
      rs += __shfl_xor(rs, 1, 16);
      rs += __shfl_xor(rs, 2, 16);
      rs += __shfl_xor(rs, 4, 16);
      rs += __shfl_xor(rs, 8, 16);
      p[j] = x[j] + rs;   // inclusive prefix within tile, C/D layout
    }
  }
#else
  {
    float myrow = 0.0f;
    if (lane < 16) {
      for (int k2 = 0; k2 < 16; ++k2) myrow += lds_t[tb + 16 * lane + k2];
    }
    float sc = myrow;  // inclusive scan of row sums across lanes 0..15
    for (int d = 1; d < 16; d <<= 1) {
      float o = __shfl_up(sc, d, 32);
      if (lane >= d && lane < 16) sc += o;
    }
#pragma unroll
    for (int j = 0; j < 8; ++j) {
      int r = j + 8 * half;
      float ro = (r == 0) ? 0.0f : __shfl(sc, r - 1, 32);
      float partial = 0.0f;
      for (int k2 = 0; k2 <= n; ++k2) partial += lds_t[tb + 16 * r + k2];
      p[j] = ro + partial;
    }
  }
#endif

  // ---- Cross-tile exclusive offsets via 32-entry LDS scan ----
  float tiletot = __shfl(p[7], 31, 32);
  if (lane == 0) lds_tot[wave] = tiletot;
  __syncthreads();
  float off;
  {
    float tv2 = (lane < wave) ? lds_tot[lane] : 0.0f;
    tv2 += __shfl_xor(tv2, 1, 32);
    tv2 += __shfl_xor(tv2, 2, 32);
    tv2 += __shfl_xor(tv2, 4, 32);
    tv2 += __shfl_xor(tv2, 8, 32);
    tv2 += __shfl_xor(tv2, 16, 32);
    off = tv2;  // sum of all previous tiles, same in every lane
  }

  // ---- Per-element contribution: log(t_i) - log(cs_i - cs_{i-1}) ----
  float re_lo[8], re_hi[8];
#pragma unroll
  for (int j = 0; j < 8; ++j) {
    re_lo[j] = __shfl(p[j], 15, 32);  // end of row j
    re_hi[j] = __shfl(p[j], 31, 32);  // end of row j+8
  }
  float prevEnd[8];
  if (half == 0) {
    prevEnd[0] = 0.0f;
    for (int j = 1; j < 8; ++j) prevEnd[j] = re_lo[j - 1];
  } else {
    prevEnd[0] = re_lo[7];
    for (int j = 1; j < 8; ++j) prevEnd[j] = re_hi[j - 1];
  }

  float acc = 0.0f;
#pragma unroll
  for (int j = 0; j < 8; ++j) {
    float left  = __shfl_up(p[j], 1, 16);
    float prevP = (n == 0) ? prevEnd[j] : left;
    int r = j + 8 * half;
    float tval = lds_t[tb + 16 * r + n];
    float cs  = off + p[j];
    float csp = off + prevP;
    acc += logf(tval) - logf(cs - csp);
  }

  // ---- Deterministic workgroup reduction -> rowsums[row] ----
  acc += __shfl_xor(acc, 1, 32);
  acc += __shfl_xor(acc, 2, 32);
  acc += __shfl_xor(acc, 4, 32);
  acc += __shfl_xor(acc, 8, 32);
  acc += __shfl_xor(acc, 16, 32);
  if (lane == 0) lds_wsum[wave] = acc;
  __syncthreads();
  if (tid == 0) {
    float s = 0.0f;
    for (int w2 = 0; w2 < NWAVES; ++w2) s += lds_wsum[w2];
    rowsums[row] = s;
  }
}

__global__ void listmle_reduce_kernel(const float* __restrict__ rowsums,
                                      float* __restrict__ out) {
  __shared__ float sh[256];
  const int tid = threadIdx.x;
  float s = 0.0f;
  for (int i = tid; i < BATCH; i += 256) s += rowsums[i];  // fixed order, deterministic
  sh[tid] = s;
  __syncthreads();
  for (int ofs = 128; ofs > 0; ofs >>= 1) {
    if (tid < ofs) sh[tid] += sh[tid + ofs];
    __syncthreads();
  }
  if (tid == 0) out[0] = -sh[0] / (float)BATCH;
}

extern "C" void kernel_launch(void* const* d_in, const int* in_sizes, int n_in,
                              void* d_out, int out_size, void* d_ws, size_t ws_size,
                              hipStream_t stream) {
  (void)in_sizes; (void)n_in; (void)out_size; (void)ws_size;
  const float* scores = (const float*)d_in[0];  // [B, N, 1] f32
  const float* labels = (const float*)d_in[1];  // [B, N]    f32
  float* rowsums = (float*)d_ws;                // BATCH floats of scratch

  (void)hipFuncSetAttribute((const void*)listmle_row_kernel,
                            hipFuncAttributeMaxDynamicSharedMemorySize, SHMEM_BYTES);

  listmle_row_kernel<<<BATCH, NTHREADS, SHMEM_BYTES, stream>>>(scores, labels, rowsums);
  listmle_reduce_kernel<<<1, 256, 0, stream>>>(rowsums, (float*)d_out);
}